// SkillRetriever_8581344657489
// MI455X (gfx1250) — compile-verified
//
#include <hip/hip_runtime.h>
#include <hip/hip_bf16.h>
#include <stdint.h>

typedef float v2f __attribute__((ext_vector_type(2)));
typedef float v8f __attribute__((ext_vector_type(8)));

#define BATCH 64
#define SEQ   1024
#define DMODEL 2048
#define SKD   128
#define NSK   65536
#define TOPK  8
#define INV_SQRT_SKD 0.08838834764831845f

// D = A(16x4, f32) * B(4x16, f32) + C(16x16, f32)
__device__ __forceinline__ v8f wmma4(v2f a, v2f b, v8f c) {
    return __builtin_amdgcn_wmma_f32_16x16x4_f32(false, a, false, b, (short)0, c,
                                                 false, false);
}

// ---------------------------------------------------------------------------
// Kernel 1: query[64,128] = h[:, -1, :] @ Wq^T + bq     (M=64, N=128, K=2048)
// grid = 32 blocks (4 M-tiles x 8 N-tiles), 32 threads (one wave) per block.
// ---------------------------------------------------------------------------
__global__ __launch_bounds__(32) void query_kernel(
    const float* __restrict__ h, const float* __restrict__ Wq,
    const float* __restrict__ bq, float* __restrict__ q_out) {
    const int mt = blockIdx.x & 3;
    const int nt = blockIdx.x >> 2;
    const int lane = threadIdx.x;
    const int lo = lane & 15, hi = lane >> 4;

    const int b = mt * 16 + lo;              // A row (M)
    const int n = nt * 16 + lo;              // B col (N)
    const float* __restrict__ arow =
        h + (size_t)b * SEQ * DMODEL + (size_t)(SEQ - 1) * DMODEL;  // last token only
    const float* __restrict__ brow = Wq + (size_t)n * DMODEL;       // Wq[n, :]

    v8f acc = {};
    for (int k = 0; k < DMODEL; k += 4) {
        const int ka = k + 2 * hi;
        v2f a  = *(const v2f*)(arow + ka);   // A[m][ka], A[m][ka+1]
        v2f bb = *(const v2f*)(brow + ka);   // B[ka][n], B[ka+1][n] (= Wq[n][ka..])
        acc = wmma4(a, bb, acc);
    }
    const float bias = bq[n];
#pragma unroll
    for (int v = 0; v < 8; ++v) {
        const int row = mt * 16 + v + 8 * hi;
        q_out[row * SKD + n] = acc[v] + bias;
    }
}

// ---------------------------------------------------------------------------
// Kernel 2: qk[64,128] = (query @ Wk) * (1/sqrt(128))   (M=64, N=128, K=128)
// scores[b,j] = qk[b] . skill[j]  + cb[b]
// ---------------------------------------------------------------------------
__global__ __launch_bounds__(32) void qk_kernel(
    const float* __restrict__ q, const float* __restrict__ Wk,
    float* __restrict__ qk_out) {
    const int mt = blockIdx.x & 3;
    const int nt = blockIdx.x >> 2;
    const int lane = threadIdx.x;
    const int lo = lane & 15, hi = lane >> 4;

    const int b = mt * 16 + lo;
    const int n = nt * 16 + lo;

    v8f acc = {};
    for (int k = 0; k < SKD; k += 4) {
        const int ka = k + 2 * hi;
        v2f a = *(const v2f*)(q + b * SKD + ka);
        v2f bb;
        bb.x = Wk[(ka + 0) * SKD + n];       // B[k][d] = Wk[k*128 + d]
        bb.y = Wk[(ka + 1) * SKD + n];
        acc = wmma4(a, bb, acc);
    }
#pragma unroll
    for (int v = 0; v < 8; ++v) {
        const int row = mt * 16 + v + 8 * hi;
        qk_out[row * SKD + n] = acc[v] * INV_SQRT_SKD;
    }
}

// cb[b] = (query[b] . bk) / sqrt(128)
__global__ void cb_kernel(const float* __restrict__ q,
                          const float* __restrict__ bk,
                          float* __restrict__ cb) {
    const int b = threadIdx.x;
    float s = 0.f;
    for (int n = 0; n < SKD; ++n) s += q[b * SKD + n] * bk[n];
    cb[b] = s * INV_SQRT_SKD;
}

// ---------------------------------------------------------------------------
// Kernel 3: scores[64, 65536] = qk @ skill^T + cb[b]
// 512 blocks x 256 threads. Each block: async-copy 128x128 skill tile to LDS
// (rows padded to 132 floats -> conflict-free ds_load_b64 B-fragments),
// 8 waves each own a 16-wide j sub-column, 4 M-tile accumulators per wave
// sharing every B fragment. 128 v_wmma per wave.
// ---------------------------------------------------------------------------
#define JT 128
#define LDS_STRIDE 132   // 128 + 4 floats pad (16B, keeps b128 alignment)

__global__ __launch_bounds__(256) void scores_kernel(
    const float* __restrict__ qk, const float* __restrict__ cb,
    const float* __restrict__ skills, float* __restrict__ scores) {
    __shared__ float tile[JT * LDS_STRIDE];

    const int t = threadIdx.x;

    // ---- async copy: 128 rows x 512B -> LDS rows of 528B --------------------
    const uint64_t gbase =
        (uint64_t)(uintptr_t)skills + (uint64_t)blockIdx.x * JT * SKD * 4;
    const unsigned ldsbase = (unsigned)(uintptr_t)(void*)tile;  // low 32 = LDS offset
#pragma unroll
    for (int i = 0; i < 16; ++i) {
        const int q16 = i * 256 + t;       // float4 index 0..4095
        const int row = q16 >> 5;          // 32 float4 per row
        const int col = q16 & 31;
        const unsigned loff = ldsbase + row * (LDS_STRIDE * 4) + col * 16;
        const unsigned goff = (unsigned)(row * (SKD * 4) + col * 16);
        asm volatile("global_load_async_to_lds_b128 %0, %1, %2"
                     :: "v"(loff), "v"(goff), "s"(gbase) : "memory");
    }
    asm volatile("s_wait_asynccnt 0" ::: "memory");
    __syncthreads();

    // ---- WMMA: M=64 (b), N=16 per wave (j), K=128 (d) -----------------------
    const int w = t >> 5;                   // wave id 0..7
    const int lane = t & 31;
    const int lo = lane & 15, hi = lane >> 4;

    const float* __restrict__ bptr = &tile[(w * 16 + lo) * LDS_STRIDE]; // skill[j][:]
    const float* __restrict__ a0 = qk + (0  + lo) * SKD;
    const float* __restrict__ a1 = qk + (16 + lo) * SKD;
    const float* __restrict__ a2 = qk + (32 + lo) * SKD;
    const float* __restrict__ a3 = qk + (48 + lo) * SKD;

    v8f c0 = {}, c1 = {}, c2 = {}, c3 = {};
    for (int k = 0; k < SKD; k += 4) {
        const int ka = k + 2 * hi;
        v2f bb = *(const v2f*)(bptr + ka);   // B[d][j] = tile[j][d..d+1]
        c0 = wmma4(*(const v2f*)(a0 + ka), bb, c0);
        c1 = wmma4(*(const v2f*)(a1 + ka), bb, c1);
        c2 = wmma4(*(const v2f*)(a2 + ka), bb, c2);
        c3 = wmma4(*(const v2f*)(a3 + ka), bb, c3);
    }

    const size_t jcol = (size_t)blockIdx.x * JT + w * 16 + lo;
#pragma unroll
    for (int v = 0; v < 8; ++v) {
        const int r = v + 8 * hi;
        scores[(size_t)(r +  0) * NSK + jcol] = c0[v] + cb[r +  0];
        scores[(size_t)(r + 16) * NSK + jcol] = c1[v] + cb[r + 16];
        scores[(size_t)(r + 32) * NSK + jcol] = c2[v] + cb[r + 32];
        scores[(size_t)(r + 48) * NSK + jcol] = c3[v] + cb[r + 48];
    }
}

// ---------------------------------------------------------------------------
// Kernel 4: per-row top-8 (jax tie-break: smaller index first), softmax over 8,
// gather skill rows, weighted sum. One block per batch row.
// out layout: [0 .. 64*128) retrieved, [64*128 .. 64*128+64*8) indices (as f32)
// ---------------------------------------------------------------------------
__device__ __forceinline__ bool better(float va, int ia, float vb, int ib) {
    return (va > vb) || (va == vb && ia < ib);
}

__global__ __launch_bounds__(256) void topk_kernel(
    const float* __restrict__ scores, const float* __restrict__ skills,
    float* __restrict__ out) {
    __shared__ float sv[256 * TOPK];
    __shared__ int   si[256 * TOPK];
    __shared__ float wts[TOPK];
    __shared__ int   fidx[TOPK];

    const int b = blockIdx.x;
    const int t = threadIdx.x;
    const float* __restrict__ row = scores + (size_t)b * NSK;

    float v[TOPK];
    int   ix[TOPK];
#pragma unroll
    for (int k = 0; k < TOPK; ++k) { v[k] = -INFINITY; ix[k] = 0x7fffffff; }

    for (int j = t; j < NSK; j += 256) {
        const float s = row[j];
        if (better(s, j, v[TOPK - 1], ix[TOPK - 1])) {
            v[TOPK - 1] = s; ix[TOPK - 1] = j;
#pragma unroll
            for (int k = TOPK - 1; k > 0; --k) {
                if (better(v[k], ix[k], v[k - 1], ix[k - 1])) {
                    float tv = v[k]; v[k] = v[k - 1]; v[k - 1] = tv;
                    int   ti = ix[k]; ix[k] = ix[k - 1]; ix[k - 1] = ti;
                }
            }
        }
    }

    // tree merge of sorted-descending 8-lists
    for (int off = 128; off >= 1; off >>= 1) {
#pragma unroll
        for (int k = 0; k < TOPK; ++k) { sv[t * TOPK + k] = v[k]; si[t * TOPK + k] = ix[k]; }
        __syncthreads();
        if (t < off) {
            float ov[TOPK]; int oi[TOPK];
#pragma unroll
            for (int k = 0; k < TOPK; ++k) {
                ov[k] = sv[(t + off) * TOPK + k];
                oi[k] = si[(t + off) * TOPK + k];
            }
            float nv[TOPK]; int ni[TOPK];
            int i = 0, j = 0;
#pragma unroll
            for (int k = 0; k < TOPK; ++k) {
                if (better(v[i], ix[i], ov[j], oi[j])) { nv[k] = v[i];  ni[k] = ix[i];  ++i; }
                else                                    { nv[k] = ov[j]; ni[k] = oi[j]; ++j; }
            }
#pragma unroll
            for (int k = 0; k < TOPK; ++k) { v[k] = nv[k]; ix[k] = ni[k]; }
        }
        __syncthreads();
    }

    if (t == 0) {
        const float mx = v[0];
        float e[TOPK], sum = 0.f;
#pragma unroll
        for (int k = 0; k < TOPK; ++k) { e[k] = __expf(v[k] - mx); sum += e[k]; }
        const float inv = 1.f / sum;
#pragma unroll
        for (int k = 0; k < TOPK; ++k) {
            wts[k] = e[k] * inv;
            fidx[k] = ix[k];
            out[BATCH * SKD + b * TOPK + k] = (float)ix[k];   // indices as f32
        }
    }
    __syncthreads();

    if (t < SKD) {
        float r = 0.f;
#pragma unroll
        for (int k = 0; k < TOPK; ++k)
            r += wts[k] * skills[(size_t)fidx[k] * SKD + t];
        out[b * SKD + t] = r;
    }
}

// ---------------------------------------------------------------------------
extern "C" void kernel_launch(void* const* d_in, const int* in_sizes, int n_in,
                              void* d_out, int out_size, void* d_ws, size_t ws_size,
                              hipStream_t stream) {
    const float* h      = (const float*)d_in[0];
    const float* skills = (const float*)d_in[1];
    const float* Wq     = (const float*)d_in[2];
    const float* bq     = (const float*)d_in[3];
    const float* Wk     = (const float*)d_in[4];
    const float* bk     = (const float*)d_in[5];
    float* out = (float*)d_out;

    float* wsf    = (float*)d_ws;
    float* q      = wsf;               // 64*128
    float* qk     = wsf + 8192;        // 64*128
    float* cb     = wsf + 16384;       // 64
    float* scores = wsf + 16448;       // 64*65536 (16 MB)

    query_kernel <<<32, 32, 0, stream>>>(h, Wq, bq, q);
    qk_kernel    <<<32, 32, 0, stream>>>(q, Wk, qk);
    cb_kernel    <<<1, 64, 0, stream>>>(q, bk, cb);
    scores_kernel<<<NSK / JT, 256, 0, stream>>>(qk, cb, skills, scores);
    topk_kernel  <<<BATCH, 256, 0, stream>>>(scores, skills, out);
}